// GraphLabelEmbedding_77979426226629
// MI455X (gfx1250) — compile-verified
//
#include <hip/hip_runtime.h>
#include <hip/hip_bf16.h>

#define NNODES 50000
#define NEDGES 800000
#define DIM    128

typedef float v2f __attribute__((ext_vector_type(2)));
typedef float v8f __attribute__((ext_vector_type(8)));

// ---------------------------------------------------------------- utilities
__global__ void zero_f32(float* __restrict__ p, int n) {
    int i = blockIdx.x * blockDim.x + threadIdx.x;
    if (i < n) p[i] = 0.0f;
}

// one thread per edge: count in/out degrees with native FP32 atomics
__global__ void degree_kernel(const int* __restrict__ src,
                              const int* __restrict__ dst,
                              float* __restrict__ degO,
                              float* __restrict__ degI, int E) {
    int e = blockIdx.x * blockDim.x + threadIdx.x;
    if (e < E) {
        unsafeAtomicAdd(&degO[src[e]], 1.0f);
        unsafeAtomicAdd(&degI[dst[e]], 1.0f);
    }
}

// deg -> (deg>0 ? rsqrt(deg) : 0), in place (deg>=1 when >0, matches ref)
__global__ void dinv_kernel(float* __restrict__ deg, int n) {
    int i = blockIdx.x * blockDim.x + threadIdx.x;
    if (i < n) {
        float v = deg[i];
        deg[i] = (v > 0.0f) ? rsqrtf(v) : 0.0f;
    }
}

// one wave32 per edge; lane handles 4 consecutive floats of the 128-wide row.
// agg[dst] += h[src] * dinv_out[src]
// Edge index is forced wave-uniform (readfirstlane) so src/dst/dinv become
// scalar (SMEM) loads and the row base is SALU math; the vector pipe only
// does the b128 feature load + 4 native global_atomic_add_f32 per lane.
__global__ void scatter_kernel(const float* __restrict__ h,
                               const float* __restrict__ dinvO,
                               const int* __restrict__ src,
                               const int* __restrict__ dst,
                               float* __restrict__ agg, int E) {
    int gid  = blockIdx.x * blockDim.x + threadIdx.x;
    int e    = __builtin_amdgcn_readfirstlane(gid >> 5);   // wave-uniform edge id
    int lane = threadIdx.x & 31;
    if (e >= E) return;                                    // uniform branch
    int s = src[e];                                        // s_load
    int d = dst[e];                                        // s_load
    float w = dinvO[s];                                    // s_load
    const float4* row = (const float4*)(h + (size_t)s * DIM);
    float4 v = row[lane];
    float* o = agg + (size_t)d * DIM + lane * 4;
    unsafeAtomicAdd(o + 0, v.x * w);
    unsafeAtomicAdd(o + 1, v.y * w);
    unsafeAtomicAdd(o + 2, v.z * w);
    unsafeAtomicAdd(o + 3, v.w * w);
}

// out[r,:] = (A[r,:] * dinvI[r]) @ W + bias, via V_WMMA_F32_16X16X4_F32.
// 256 threads = 8 waves; block computes 128 rows x 128 cols.
// W is staged in LDS k-pair interleaved:  Wl2[kp*128 + n] = {W[2kp][n], W[2kp+1][n]}
// so each B fragment (VGPR0 = row kk, VGPR1 = row kk+1 for this lane's half)
// is one aligned ds_load_b64 — no register shuffling before the WMMA.
__global__ void __launch_bounds__(256)
gcn_gemm_kernel(const float* __restrict__ A,
                const float* __restrict__ dinvI,
                const float* __restrict__ W,
                const float* __restrict__ bias,
                float* __restrict__ out, int nRows) {
    __shared__ float Wl[DIM * DIM];   // 64 KB, pair-interleaved
    {
        v2f* Wl2 = (v2f*)Wl;
        for (int i = threadIdx.x; i < (DIM / 2) * DIM; i += blockDim.x) {
            int kp = i >> 7;          // k-pair index 0..63
            int n  = i & 127;         // column
            v2f p;
            p.x = W[(size_t)(2 * kp)     * DIM + n];
            p.y = W[(size_t)(2 * kp + 1) * DIM + n];
            Wl2[i] = p;               // i == kp*128 + n
        }
    }
    __syncthreads();

    const int lane = threadIdx.x & 31;
    const int wave = threadIdx.x >> 5;
    const int hi   = lane >> 4;     // half-wave: K pair (2kb) vs (2kb+1)
    const int lcol = lane & 15;

    const int rowBase = blockIdx.x * 128 + wave * 16;
    const int row     = rowBase + lcol;          // A-fragment row for this lane
    const int r       = (row < nRows) ? row : (nRows - 1);  // clamp tail (store-masked)
    const float dsc   = dinvI[r];
    const float* arow = A + (size_t)r * DIM;

    const v2f* Wp = (const v2f*)Wl;

    v8f acc[8];
    v8f zero = {};
#pragma unroll
    for (int nt = 0; nt < 8; ++nt) acc[nt] = zero;

    for (int kb = 0; kb < DIM / 4; ++kb) {
        const int kk = kb * 4 + hi * 2;          // this lane's K pair: kk, kk+1
        v2f a;
        a.x = arow[kk]     * dsc;
        a.y = arow[kk + 1] * dsc;
        const int kp   = kb * 2 + hi;            // LDS k-pair row
        const int base = kp * 128 + lcol;        // v2f units
#pragma unroll
        for (int nt = 0; nt < 8; ++nt) {
            v2f bf = Wp[base + nt * 16];         // single ds_load_b64
            acc[nt] = __builtin_amdgcn_wmma_f32_16x16x4_f32(
                false, a, false, bf, (short)0, acc[nt], false, false);
        }
    }

    float bcol[8];
#pragma unroll
    for (int nt = 0; nt < 8; ++nt) bcol[nt] = bias[nt * 16 + lcol];

#pragma unroll
    for (int nt = 0; nt < 8; ++nt) {
#pragma unroll
        for (int v = 0; v < 8; ++v) {
            int m    = v + hi * 8;               // D-tile row within 16
            int orow = rowBase + m;
            if (orow < nRows)
                out[(size_t)orow * DIM + nt * 16 + lcol] = acc[nt][v] + bcol[nt];
        }
    }
}

// out[i,:] = h[labels[i],:]
__global__ void gather_kernel(const float* __restrict__ h,
                              const int* __restrict__ labels,
                              float* __restrict__ out, int nLab) {
    int i = blockIdx.x * blockDim.x + threadIdx.x;
    if (i < nLab * DIM) {
        int r = i >> 7;
        int c = i & (DIM - 1);
        out[i] = h[(size_t)labels[r] * DIM + c];
    }
}

// ---------------------------------------------------------------- launcher
extern "C" void kernel_launch(void* const* d_in, const int* in_sizes, int n_in,
                              void* d_out, int out_size, void* d_ws, size_t ws_size,
                              hipStream_t stream) {
    const int*   labels = (const int*)d_in[0];
    const int*   src    = (const int*)d_in[1];
    const int*   dst    = (const int*)d_in[2];
    const float* feat   = (const float*)d_in[3];
    const float* W0     = (const float*)d_in[4];
    const float* b0     = (const float*)d_in[5];
    const float* W1     = (const float*)d_in[6];
    const float* b1     = (const float*)d_in[7];
    const float* W2     = (const float*)d_in[8];
    const float* b2     = (const float*)d_in[9];
    float* outp = (float*)d_out;

    const int NF = NNODES * DIM;  // 6,400,000 floats per node buffer

    float* ws    = (float*)d_ws;
    float* dinvO = ws;                 // 50000
    float* dinvI = ws + NNODES;        // 50000   (contiguous with dinvO)
    float* agg   = ws + 2 * NNODES;    // NF  (16B aligned: 100000*4 % 16 == 0)
    float* hA    = agg + NF;           // NF
    float* hB    = hA + NF;            // NF

    const int T = 256;

    // degrees -> dinv (both arrays handled as one contiguous 100000 run)
    zero_f32<<<(2 * NNODES + T - 1) / T, T, 0, stream>>>(dinvO, 2 * NNODES);
    degree_kernel<<<(NEDGES + T - 1) / T, T, 0, stream>>>(src, dst, dinvO, dinvI, NEDGES);
    dinv_kernel<<<(2 * NNODES + T - 1) / T, T, 0, stream>>>(dinvO, 2 * NNODES);

    const int scatterBlocks = (NEDGES * 32 + T - 1) / T;   // one wave per edge
    const int gemmBlocks    = (NNODES + 127) / 128;        // 391

    // layer 1: feat -> hA
    zero_f32<<<(NF + T - 1) / T, T, 0, stream>>>(agg, NF);
    scatter_kernel<<<scatterBlocks, T, 0, stream>>>(feat, dinvO, src, dst, agg, NEDGES);
    gcn_gemm_kernel<<<gemmBlocks, T, 0, stream>>>(agg, dinvI, W0, b0, hA, NNODES);

    // layer 2: hA -> hB
    zero_f32<<<(NF + T - 1) / T, T, 0, stream>>>(agg, NF);
    scatter_kernel<<<scatterBlocks, T, 0, stream>>>(hA, dinvO, src, dst, agg, NEDGES);
    gcn_gemm_kernel<<<gemmBlocks, T, 0, stream>>>(agg, dinvI, W1, b1, hB, NNODES);

    // layer 3: hB -> hA
    zero_f32<<<(NF + T - 1) / T, T, 0, stream>>>(agg, NF);
    scatter_kernel<<<scatterBlocks, T, 0, stream>>>(hB, dinvO, src, dst, agg, NEDGES);
    gcn_gemm_kernel<<<gemmBlocks, T, 0, stream>>>(agg, dinvI, W2, b2, hA, NNODES);

    // label gather
    gather_kernel<<<(4096 * DIM + T - 1) / T, T, 0, stream>>>(hA, labels, outp, 4096);
}